// H3_50603304681584
// MI455X (gfx1250) — compile-verified
//
#include <hip/hip_runtime.h>

typedef __attribute__((ext_vector_type(2))) float v2f;
typedef __attribute__((ext_vector_type(8))) float v8f;

#define B_SZ    8
#define D_MODEL 512
#define L_SEQ   2048
#define N2      32
#define DSTATE  64
#define NROW    (B_SZ * D_MODEL)
#define NELEM   ((size_t)B_SZ * D_MODEL * L_SEQ)
#define XPITCH  33  // padded LDS row pitch (dwords) -> conflict-free banks

#if __has_builtin(__builtin_amdgcn_global_load_async_to_lds_b32) && \
    __has_builtin(__builtin_amdgcn_s_wait_asynccnt)
#define USE_ASYNC_LDS 1
typedef __attribute__((address_space(1))) int* gptr_i32;
typedef __attribute__((address_space(3))) int* lptr_i32;
#else
#define USE_ASYNC_LDS 0
#endif

// ---------------------------------------------------------------------------
// K1: fused q/k/v projection GEMM via V_WMMA_F32_16X16X4_F32.
// Grid: (L/32, D/128, B); block 256 = 8 waves; wave w owns a 16x32 output tile
// (rows mblk*128 + w*16, cols lblk*32) for all three projections, so each W
// row-slice is reused across two N tiles (halves L2 traffic on the W stream).
// A (16x4 f32): lane -> M = lane&15, VGPR0/1 = K{0,1} (lanes 0-15) / K{2,3}.
// B (4x16 f32): lane -> N = lane&15, VGPR0/1 = K{0,1}/{2,3} by lane half.
// C (16x16 f32): VGPR r -> M = r (lanes 0-15) / r+8 (lanes 16-31), N = lane&15.
// X tile staged into LDS *transposed* (Xs[n][kk], pitch 33) so each B operand's
// K-pair is contiguous -> single ds_load_2addr into a consecutive VGPR pair,
// no repack movs. Staging uses GLOBAL_LOAD_ASYNC_TO_LDS_B32 (ASYNCcnt path).
// ---------------------------------------------------------------------------
__global__ __launch_bounds__(256)
void h3_qkv_gemm(const float* __restrict__ x,
                 const float* __restrict__ Wq, const float* __restrict__ bq,
                 const float* __restrict__ Wk, const float* __restrict__ bk,
                 const float* __restrict__ Wv, const float* __restrict__ bv,
                 float* __restrict__ qo, float* __restrict__ ko,
                 float* __restrict__ vo) {
  __shared__ float Xs[32 * XPITCH];  // [n=0..31][kk=0..31], padded
  const int tid   = threadIdx.x;
  const int lane  = tid & 31;
  const int wave  = tid >> 5;
  const int b     = blockIdx.z;
  const int mbase = blockIdx.y * 128 + wave * 16;
  const int lbase = blockIdx.x * 32;
  const int mrow  = mbase + (lane & 15);
  const int koff  = (lane >> 4) << 1;  // 0 or 2
  const int n     = lane & 15;
  const float* xb = x + (size_t)b * D_MODEL * L_SEQ;

  v8f cq0 = {}, ck0 = {}, cv0 = {};
  v8f cq1 = {}, ck1 = {}, cv1 = {};

  for (int kc = 0; kc < D_MODEL; kc += 32) {
    __syncthreads();
    // stage X[kc..kc+31, lbase..lbase+31] into LDS transposed (1024 elems)
#if USE_ASYNC_LDS
    for (int e = tid; e < 1024; e += 256) {
      const int kk = e >> 5, nn = e & 31;
      __builtin_amdgcn_global_load_async_to_lds_b32(
          (gptr_i32)(xb + (size_t)(kc + kk) * L_SEQ + lbase + nn),
          (lptr_i32)(Xs + nn * XPITCH + kk), 0, 0);
    }
    __builtin_amdgcn_s_wait_asynccnt(0);
#else
    for (int e = tid; e < 1024; e += 256) {
      const int kk = e >> 5, nn = e & 31;
      Xs[nn * XPITCH + kk] = xb[(size_t)(kc + kk) * L_SEQ + lbase + nn];
    }
#endif
    __syncthreads();
    if (kc + 32 < D_MODEL) {
      __builtin_prefetch(Wq + (size_t)mrow * D_MODEL + kc + 32 + koff, 0, 0);
      __builtin_prefetch(Wk + (size_t)mrow * D_MODEL + kc + 32 + koff, 0, 0);
      __builtin_prefetch(Wv + (size_t)mrow * D_MODEL + kc + 32 + koff, 0, 0);
    }
#pragma unroll
    for (int kk = 0; kk < 32; kk += 4) {
      v2f bm0, bm1;
      bm0.x = Xs[n * XPITCH + kk + koff];
      bm0.y = Xs[n * XPITCH + kk + koff + 1];
      bm1.x = Xs[(n + 16) * XPITCH + kk + koff];
      bm1.y = Xs[(n + 16) * XPITCH + kk + koff + 1];
      const v2f aq = *(const v2f*)(Wq + (size_t)mrow * D_MODEL + kc + kk + koff);
      const v2f ak = *(const v2f*)(Wk + (size_t)mrow * D_MODEL + kc + kk + koff);
      const v2f av = *(const v2f*)(Wv + (size_t)mrow * D_MODEL + kc + kk + koff);
      cq0 = __builtin_amdgcn_wmma_f32_16x16x4_f32(false, aq, false, bm0,
                                                  (short)0, cq0, false, false);
      cq1 = __builtin_amdgcn_wmma_f32_16x16x4_f32(false, aq, false, bm1,
                                                  (short)0, cq1, false, false);
      ck0 = __builtin_amdgcn_wmma_f32_16x16x4_f32(false, ak, false, bm0,
                                                  (short)0, ck0, false, false);
      ck1 = __builtin_amdgcn_wmma_f32_16x16x4_f32(false, ak, false, bm1,
                                                  (short)0, ck1, false, false);
      cv0 = __builtin_amdgcn_wmma_f32_16x16x4_f32(false, av, false, bm0,
                                                  (short)0, cv0, false, false);
      cv1 = __builtin_amdgcn_wmma_f32_16x16x4_f32(false, av, false, bm1,
                                                  (short)0, cv1, false, false);
    }
  }

  const int half    = (lane >> 4) << 3;
  const size_t boff = (size_t)b * D_MODEL * L_SEQ;
#pragma unroll
  for (int r = 0; r < 8; ++r) {
    const int m    = mbase + r + half;
    const size_t o = boff + (size_t)m * L_SEQ + lbase + n;
    const float biq = bq[m], bik = bk[m], biv = bv[m];
    qo[o]      = cq0[r] + biq;
    qo[o + 16] = cq1[r] + biq;
    ko[o]      = ck0[r] + bik;
    ko[o + 16] = ck1[r] + bik;
    vo[o]      = cv0[r] + biv;
    vo[o + 16] = cv1[r] + biv;
  }
}

// ---------------------------------------------------------------------------
// K2: shift SSM (causal 64-tap FIR with taps shift_C[h,0:64]) + k*shift_D,
// fused with the elementwise u = v * shift_out.
// One block per (row, 256-wide l-chunk).
// ---------------------------------------------------------------------------
__global__ __launch_bounds__(256)
void h3_shift_mul(const float* __restrict__ kin, const float* __restrict__ vin,
                  const float* __restrict__ shiftC, const float* __restrict__ shiftD,
                  float* __restrict__ uo) {
  __shared__ float taps[DSTATE];
  __shared__ float kseg[320];  // 63 halo + 256 + pad
  const int t     = threadIdx.x;
  const int row   = blockIdx.x >> 3;
  const int lbase = (blockIdx.x & 7) * 256;
  const int h     = row & (D_MODEL - 1);
  const float* krow = kin + (size_t)row * L_SEQ;

  if (t < DSTATE) taps[t] = shiftC[h * DSTATE + t];
  {
    int i  = t;
    int gl = lbase + i - 63;
    kseg[i] = (gl >= 0) ? krow[gl] : 0.0f;
    i = t + 256;
    if (i < 319) {
      gl = lbase + i - 63;
      kseg[i] = (gl >= 0) ? krow[gl] : 0.0f;
    }
  }
  __syncthreads();

  float acc = shiftD[h] * kseg[t + 63];
#pragma unroll
  for (int s = 0; s < DSTATE; ++s)
    acc = fmaf(taps[s], kseg[t + 63 - s], acc);

  const size_t o = (size_t)row * L_SEQ + lbase + t;
  uo[o] = vin[o] * acc;
}

// ---------------------------------------------------------------------------
// K3: S4D discretization. Per (h,n): w = exp(dt*A), G = 2*C*(w-1)/A.
// ---------------------------------------------------------------------------
__global__ void h3_s4d_coeffs(const float* __restrict__ log_dt,
                              const float* __restrict__ A_real,
                              const float* __restrict__ A_imag,
                              const float* __restrict__ C_re,
                              const float* __restrict__ C_im,
                              float* __restrict__ wr, float* __restrict__ wi,
                              float* __restrict__ gr, float* __restrict__ gi) {
  const int idx = blockIdx.x * blockDim.x + threadIdx.x;
  if (idx >= D_MODEL * N2) return;
  const int h = idx >> 5;
  const float dt = __expf(log_dt[h]);
  const float Ar = -__expf(A_real[idx]);
  const float Ai = A_imag[idx];
  const float ar = Ar * dt, ai = Ai * dt;
  const float er = __expf(ar);
  float s, c;
  __sincosf(ai, &s, &c);
  const float wwr = er * c, wwi = er * s;
  const float inv = 1.0f / (Ar * Ar + Ai * Ai);
  const float pr  = ((wwr - 1.0f) * Ar + wwi * Ai) * inv;  // (w-1)/A
  const float pi  = (wwi * Ar - (wwr - 1.0f) * Ai) * inv;
  const float cr  = C_re[idx], ci = C_im[idx];
  wr[idx] = wwr;
  wi[idx] = wwi;
  gr[idx] = 2.0f * (cr * pr - ci * pi);
  gi[idx] = 2.0f * (cr * pi + ci * pr);
}

// ---------------------------------------------------------------------------
// K4: S4D diagonal scan. One wave per (b,h) row; lane = complex mode.
// Per step: complex FMA state update, butterfly mode-sum, lane j keeps y[t0+j].
// Fused with y += u*s4_D and the final q* multiply.
// ---------------------------------------------------------------------------
__global__ __launch_bounds__(256)
void h3_s4d_scan(const float* __restrict__ u, const float* __restrict__ q,
                 const float* __restrict__ wr, const float* __restrict__ wi,
                 const float* __restrict__ gr, const float* __restrict__ gi,
                 const float* __restrict__ s4D, float* __restrict__ out) {
  const int lane = threadIdx.x & 31;
  const int wave = threadIdx.x >> 5;
  const int row  = blockIdx.x * 8 + wave;
  const int h    = row & (D_MODEL - 1);
  const float W_r = wr[h * N2 + lane];
  const float W_i = wi[h * N2 + lane];
  const float G_r = gr[h * N2 + lane];
  const float G_i = gi[h * N2 + lane];
  const float Dp  = s4D[h];
  const float* urow = u + (size_t)row * L_SEQ;
  const float* qrow = q + (size_t)row * L_SEQ;
  float* orow       = out + (size_t)row * L_SEQ;

  float Sr = 0.0f, Si = 0.0f;
  for (int t0 = 0; t0 < L_SEQ; t0 += 32) {
    const float uvec = urow[t0 + lane];
    float yl = 0.0f;
#pragma unroll
    for (int j = 0; j < 32; ++j) {
      const float ub  = __shfl(uvec, j, 32);
      const float nSr = fmaf(W_r, Sr, fmaf(-W_i, Si, ub));
      const float nSi = fmaf(W_r, Si, W_i * Sr);
      Sr = nSr;
      Si = nSi;
      float cc = fmaf(G_r, Sr, -G_i * Si);
#pragma unroll
      for (int m = 16; m >= 1; m >>= 1) cc += __shfl_xor(cc, m, 32);
      yl = (lane == j) ? cc : yl;
    }
    const float qv = qrow[t0 + lane];
    orow[t0 + lane] = qv * fmaf(Dp, uvec, yl);
  }
}

// ---------------------------------------------------------------------------
extern "C" void kernel_launch(void* const* d_in, const int* in_sizes, int n_in,
                              void* d_out, int out_size, void* d_ws, size_t ws_size,
                              hipStream_t stream) {
  (void)in_sizes; (void)n_in; (void)out_size; (void)ws_size;
  const float* x      = (const float*)d_in[0];
  const float* Wq     = (const float*)d_in[1];
  const float* bq     = (const float*)d_in[2];
  const float* Wk     = (const float*)d_in[3];
  const float* bk     = (const float*)d_in[4];
  const float* Wv     = (const float*)d_in[5];
  const float* bv     = (const float*)d_in[6];
  const float* sC     = (const float*)d_in[7];
  const float* sD     = (const float*)d_in[8];
  const float* log_dt = (const float*)d_in[9];
  const float* A_real = (const float*)d_in[10];
  const float* A_imag = (const float*)d_in[11];
  const float* C_re   = (const float*)d_in[12];
  const float* C_im   = (const float*)d_in[13];
  const float* s4D    = (const float*)d_in[14];
  float* out = (float*)d_out;

  float* ws   = (float*)d_ws;
  float* q    = ws;
  float* k    = ws + NELEM;
  float* v    = ws + 2 * NELEM;
  float* u    = ws + 3 * NELEM;
  float* coef = ws + 4 * NELEM;
  float* wr = coef;
  float* wi = coef + D_MODEL * N2;
  float* gr = coef + 2 * D_MODEL * N2;
  float* gi = coef + 3 * D_MODEL * N2;

  dim3 g1(L_SEQ / 32, D_MODEL / 128, B_SZ);
  h3_qkv_gemm<<<g1, 256, 0, stream>>>(x, Wq, bq, Wk, bk, Wv, bv, q, k, v);
  h3_shift_mul<<<NROW * (L_SEQ / 256), 256, 0, stream>>>(k, v, sC, sD, u);
  h3_s4d_coeffs<<<(D_MODEL * N2 + 255) / 256, 256, 0, stream>>>(
      log_dt, A_real, A_imag, C_re, C_im, wr, wi, gr, gi);
  h3_s4d_scan<<<NROW / 8, 256, 0, stream>>>(u, q, wr, wi, gr, gi, s4D, out);
}